// LRU_12575664242949
// MI455X (gfx1250) — compile-verified
//
#include <hip/hip_runtime.h>

// Problem constants (from reference setup_inputs):
// x: [B=16, T=4096, D=512] f32, W: [D=512] f32, out: [B, T, D] f32.
// Recurrence per channel d: a = 1 + (exp(-exp(W)) - 1), b = sqrt(1 - w*w),
//   h_t = a*h_{t-1} + b*x_t, h_{-1} = 0, output is h_t sequence.
constexpr int Bn = 16;
constexpr int T  = 4096;
constexpr int D  = 512;
constexpr int C  = 32;          // chunks along T
constexpr int L  = T / C;       // 128 timesteps per chunk
constexpr int TS = 8;           // timesteps per LDS tile
constexpr int NT = L / TS;      // 16 tiles per chunk
// One tile = TS*D*4 = 16 KB. 512 threads * 16 B (b128) = 8 KB per issue -> 2 issues/tile.

__device__ __forceinline__ float chan_decay(float Wd) {
    return 1.0f + (__expf(-__expf(Wd)) - 1.0f);          // a = 1 + dt*w, dt = 1
}
__device__ __forceinline__ float chan_inscale(float Wd) {
    float w = __expf(-__expf(Wd)) - 1.0f;
    return sqrtf(1.0f - w * w);                          // b = dt*w_res
}

// Issue one 16 KB tile (TS rows x D channels) as async global->LDS copies.
// ldsOff is the workgroup-relative LDS byte offset of the destination buffer
// (low 32 bits of the generic pointer per the CDNA5 LDS aperture mapping).
__device__ __forceinline__ void async_tile_load(const float* gsrc, unsigned ldsOff) {
    unsigned long long ga = (unsigned long long)(size_t)gsrc
                          + (unsigned long long)(threadIdx.x * 16u);
    unsigned la = ldsOff + threadIdx.x * 16u;
    asm volatile("global_load_async_to_lds_b128 %0, %1, off"
                 :: "v"(la), "v"(ga) : "memory");
    asm volatile("global_load_async_to_lds_b128 %0, %1, off"
                 :: "v"((unsigned)(la + 8192u)), "v"(ga + 8192ull) : "memory");
}

// ---------------- Phase 1: per-chunk carry (h at chunk end, h0 = 0) ----------
__global__ __launch_bounds__(D) void lru_carry_kernel(
        const float* __restrict__ x, const float* __restrict__ W,
        float* __restrict__ carry) {
    __shared__ float lds[2][TS * D];
    const int blk = blockIdx.x;
    const int b = blk / C, k = blk % C;
    const int d = threadIdx.x;

    const float Wd = W[d];
    const float a  = chan_decay(Wd);
    const float bs = chan_inscale(Wd);

    const float* xbase = x + (size_t)(b * T + k * L) * D;
    const unsigned lbase0 = (unsigned)(size_t)(&lds[0][0]);
    const unsigned lbase1 = (unsigned)(size_t)(&lds[1][0]);

    float h = 0.0f;
    async_tile_load(xbase, lbase0);
    for (int j = 0; j < NT; ++j) {
        const int buf = j & 1;
        if (j + 1 < NT) {
            async_tile_load(xbase + (size_t)(j + 1) * TS * D, buf ? lbase0 : lbase1);
            asm volatile("s_wait_asynccnt 2" ::: "memory");   // tile j landed (this wave)
        } else {
            asm volatile("s_wait_asynccnt 0" ::: "memory");
        }
        __syncthreads();                                      // tile j landed (all waves)
#pragma unroll
        for (int r = 0; r < TS; ++r)
            h = fmaf(a, h, bs * lds[buf][r * D + d]);
        __syncthreads();                                      // done reading buf before refill
    }
    carry[(size_t)(b * C + k) * D + d] = h;
}

// ---------------- Phase 2: sequential prefix over C=32 chunk carries ---------
__global__ __launch_bounds__(D) void lru_prefix_kernel(
        const float* __restrict__ W, const float* __restrict__ carry,
        float* __restrict__ prefix) {
    const int b = blockIdx.x;
    const int d = threadIdx.x;
    const float a = chan_decay(W[d]);
    float aL = a;
#pragma unroll
    for (int i = 0; i < 7; ++i) aL *= aL;                     // a^128 = a^L
    float H = 0.0f;                                           // state at start of chunk 0
    for (int k = 0; k < C; ++k) {
        const size_t idx = (size_t)(b * C + k) * D + d;
        prefix[idx] = H;
        H = fmaf(aL, H, carry[idx]);
    }
}

// ---------------- Phase 3: full scan per chunk with correct initial state ----
// x is 128 MB < 192 MB L2, so this second read of x mostly hits L2.
__global__ __launch_bounds__(D) void lru_scan_kernel(
        const float* __restrict__ x, const float* __restrict__ W,
        const float* __restrict__ prefix, float* __restrict__ out) {
    __shared__ float lds[2][TS * D];
    const int blk = blockIdx.x;
    const int b = blk / C, k = blk % C;
    const int d = threadIdx.x;

    const float Wd = W[d];
    const float a  = chan_decay(Wd);
    const float bs = chan_inscale(Wd);

    const float* xbase = x   + (size_t)(b * T + k * L) * D;
    float*       ybase = out + (size_t)(b * T + k * L) * D;
    const unsigned lbase0 = (unsigned)(size_t)(&lds[0][0]);
    const unsigned lbase1 = (unsigned)(size_t)(&lds[1][0]);

    float h = prefix[(size_t)(b * C + k) * D + d];
    async_tile_load(xbase, lbase0);
    for (int j = 0; j < NT; ++j) {
        const int buf = j & 1;
        if (j + 1 < NT) {
            async_tile_load(xbase + (size_t)(j + 1) * TS * D, buf ? lbase0 : lbase1);
            asm volatile("s_wait_asynccnt 2" ::: "memory");
        } else {
            asm volatile("s_wait_asynccnt 0" ::: "memory");
        }
        __syncthreads();
        float* yrow = ybase + (size_t)j * TS * D + d;
#pragma unroll
        for (int r = 0; r < TS; ++r) {
            h = fmaf(a, h, bs * lds[buf][r * D + d]);
            yrow[r * D] = h;
        }
        __syncthreads();
    }
}

extern "C" void kernel_launch(void* const* d_in, const int* in_sizes, int n_in,
                              void* d_out, int out_size, void* d_ws, size_t ws_size,
                              hipStream_t stream) {
    (void)in_sizes; (void)n_in; (void)out_size; (void)ws_size;
    const float* x = (const float*)d_in[0];   // [B, T, D]
    const float* W = (const float*)d_in[1];   // [D]
    float* out = (float*)d_out;               // [B, T, D]

    float* carry  = (float*)d_ws;                       // B*C*D floats = 1 MB
    float* prefix = carry + (size_t)Bn * C * D;         // B*C*D floats = 1 MB

    lru_carry_kernel <<<Bn * C, D, 0, stream>>>(x, W, carry);
    lru_prefix_kernel<<<Bn,     D, 0, stream>>>(W, carry, prefix);
    lru_scan_kernel  <<<Bn * C, D, 0, stream>>>(x, W, prefix, out);
}